// Voxelization_83537113907663
// MI455X (gfx1250) — compile-verified
//
#include <hip/hip_runtime.h>
#include <stdint.h>

// ---------------- Problem constants (mirror reference) ----------------
#define GXC 1408
#define GYC 1600
#define GZC 40
#define MAXV 60000
#define MAXP 64

// Open-addressed hash table: ~2M distinct voxels into 4.19M slots (load ~0.47)
#define HBITS 22
#define HSIZE (1 << HBITS)
#define HMASK (HSIZE - 1)

// Prefix-scan tiling
#define SCAN_T 256
#define SCAN_ITEMS 16
#define SCAN_CHUNK (SCAN_T * SCAN_ITEMS)

#define HAS_ASYNC_LDS (__has_builtin(__builtin_amdgcn_global_load_async_to_lds_b32))
#if !HAS_ASYNC_LDS
#warning "gfx1250 async-to-LDS builtin NOT available; falling back to plain LDS staging"
#endif

__device__ __forceinline__ unsigned hash_lin(int lin) {
  return ((unsigned)lin * 2654435761u) >> (32 - HBITS);
}

// ---------------- K0a: zero output with b128 stores ----------------
__global__ void vox_zero_f(float* __restrict__ p, int nElems) {
  int n4 = nElems >> 2;
  float4 z = make_float4(0.f, 0.f, 0.f, 0.f);
  for (int i = blockIdx.x * blockDim.x + threadIdx.x; i < n4;
       i += gridDim.x * blockDim.x)
    ((float4*)p)[i] = z;
  // tail (<4 elems)
  int tail = n4 << 2;
  if (blockIdx.x == 0) {
    int ti = tail + threadIdx.x;
    if (ti < nElems) p[ti] = 0.0f;
  }
}

// ---------------- K0b: init hash + arrival with b128 stores ----------------
__global__ void vox_init_ws(int4* __restrict__ hKey4, int4* __restrict__ hMin4,
                            int4* __restrict__ hCnt4, int4* __restrict__ hRank4,
                            int4* __restrict__ arrival4) {
  int i = blockIdx.x * blockDim.x + threadIdx.x;
  const int4 neg1 = make_int4(-1, -1, -1, -1);
  const int4 maxi = make_int4(0x7FFFFFFF, 0x7FFFFFFF, 0x7FFFFFFF, 0x7FFFFFFF);
  const int4 zero = make_int4(0, 0, 0, 0);
  if (i < (HSIZE / 4)) {
    hKey4[i] = neg1;
    hMin4[i] = maxi;
    hCnt4[i] = zero;
    hRank4[i] = neg1;
  }
  if (i < (MAXV / 4)) arrival4[i] = zero;
}

// ---------------- K1: grid coords + hash insert ----------------
__global__ void vox_assign(const float* __restrict__ pts, int n,
                           int* __restrict__ hKey, int* __restrict__ hMin,
                           int* __restrict__ hCnt, int* __restrict__ linA,
                           int* __restrict__ slotA) {
  int i = blockIdx.x * blockDim.x + threadIdx.x;
  if (i >= n) return;
  // CDNA5 prefetch path (lowers to global_prefetch_b8)
  if (i + 8192 < n) __builtin_prefetch(pts + (size_t)(i + 8192) * 4, 0, 1);

  float4 p = ((const float4*)pts)[i];
  float fx = floorf((p.x - 0.0f)  / 0.05f);
  float fy = floorf((p.y + 40.0f) / 0.05f);
  float fz = floorf((p.z + 3.0f)  / 0.1f);
  bool valid = (fx >= 0.0f) && (fx < (float)GXC) &&
               (fy >= 0.0f) && (fy < (float)GYC) &&
               (fz >= 0.0f) && (fz < (float)GZC);
  if (!valid) {
    linA[i] = -1;
    slotA[i] = -1;
    return;
  }
  int gx = (int)fx, gy = (int)fy, gz = (int)fz;
  int lin = (gx * GYC + gy) * GZC + gz;
  linA[i] = lin;

  unsigned s = hash_lin(lin) & HMASK;
  while (true) {
    int prev = atomicCAS(&hKey[s], -1, lin);
    if (prev == -1 || prev == lin) break;
    s = (s + 1) & HMASK;
  }
  atomicMin(&hMin[s], i);
  atomicAdd(&hCnt[s], 1);
  slotA[i] = (int)s;
}

// ---------------- K2a: per-block first-occurrence counts ----------------
__global__ void vox_blocksum(const int* __restrict__ slotA,
                             const int* __restrict__ hMin, int n,
                             int* __restrict__ bsums) {
  __shared__ int sh[SCAN_T];
  int b = blockIdx.x, t = threadIdx.x;
  int base = b * SCAN_CHUNK + t * SCAN_ITEMS;
  int c = 0;
  for (int k = 0; k < SCAN_ITEMS; k++) {
    int i = base + k;
    if (i < n) {
      int s = slotA[i];
      if (s >= 0 && hMin[s] == i) c++;
    }
  }
  sh[t] = c;
  __syncthreads();
  for (int off = SCAN_T / 2; off > 0; off >>= 1) {
    if (t < off) sh[t] += sh[t + off];
    __syncthreads();
  }
  if (t == 0) bsums[b] = sh[0];
}

// ---------------- K2b: tiny serial exclusive scan (<=~500 elems) --------
__global__ void vox_scan(int nblocks, int* __restrict__ bsums,
                         int* __restrict__ misc, float* __restrict__ outVn) {
  int run = 0;
  for (int b = 0; b < nblocks; b++) {
    int v = bsums[b];
    bsums[b] = run;
    run += v;
  }
  int vn = run < MAXV ? run : MAXV;
  misc[0] = vn;
  *outVn = (float)vn;
}

// ---------------- K2c: deterministic rank assignment ----------------
__global__ void vox_rank(const int* __restrict__ slotA,
                         const int* __restrict__ linA,
                         const int* __restrict__ hMin,
                         const int* __restrict__ hCnt, int n,
                         const int* __restrict__ bsums, int* __restrict__ hRank,
                         int* __restrict__ rankSlot,
                         float* __restrict__ outCoors,
                         float* __restrict__ outNpv) {
  __shared__ int sh[SCAN_T];
  int b = blockIdx.x, t = threadIdx.x;
  int base = b * SCAN_CHUNK + t * SCAN_ITEMS;
  int flags[SCAN_ITEMS];
  int c = 0;
  for (int k = 0; k < SCAN_ITEMS; k++) {
    int i = base + k;
    int f = 0;
    if (i < n) {
      int s = slotA[i];
      if (s >= 0 && hMin[s] == i) f = 1;
    }
    flags[k] = f;
    c += f;
  }
  sh[t] = c;
  __syncthreads();
  // Hillis-Steele inclusive scan across the block
  for (int off = 1; off < SCAN_T; off <<= 1) {
    int v = 0;
    if (t >= off) v = sh[t - off];
    __syncthreads();
    sh[t] += v;
    __syncthreads();
  }
  int pre = sh[t] - c + bsums[b];  // exclusive global rank base for this thread
  for (int k = 0; k < SCAN_ITEMS; k++) {
    if (flags[k]) {
      int r = pre++;
      if (r < MAXV) {
        int i = base + k;
        int s = slotA[i];
        hRank[s] = r;
        rankSlot[r] = s;
        int lin = linA[i];
        int gz = lin % GZC;
        int rem = lin / GZC;
        int gy = rem % GYC;
        int gx = rem / GYC;
        outCoors[r * 3 + 0] = (float)gz;  // reference flips to (z,y,x)
        outCoors[r * 3 + 1] = (float)gy;
        outCoors[r * 3 + 2] = (float)gx;
        int cnt = hCnt[s];
        outNpv[r] = (float)(cnt < MAXP ? cnt : MAXP);
      }
    }
  }
}

// ---------------- K3: collect point indices per kept voxel ----------------
__global__ void vox_collect(const int* __restrict__ slotA,
                            const int* __restrict__ hRank, int n,
                            int* __restrict__ arrival, int* __restrict__ idxbuf) {
  int i = blockIdx.x * blockDim.x + threadIdx.x;
  if (i >= n) return;
  int s = slotA[i];
  if (s < 0) return;
  int r = hRank[s];
  if (r < 0) return;
  int a = atomicAdd(&arrival[r], 1);
  if (a < MAXP) idxbuf[r * MAXP + a] = i;
}

// ---------------- K4: sort indices per voxel, emit points ----------------
__global__ void vox_emit(const float* __restrict__ pts, int n,
                         const int* __restrict__ idxbuf,
                         const int* __restrict__ rankSlot,
                         const int* __restrict__ hCnt,
                         const int* __restrict__ hKey,
                         const int* __restrict__ linA,
                         const int* __restrict__ misc,
                         float* __restrict__ outVox) {
  __shared__ int shIdx[MAXP];
  __shared__ int shFlags[MAXP];
  __shared__ int shFound;
  int r = blockIdx.x, t = threadIdx.x;
  if (r >= misc[0]) return;  // inactive rank: voxels already zeroed
  int s = rankSlot[r];
  int cnt = hCnt[s];

  if (cnt <= MAXP) {
    // Stage only the cnt valid (arbitrary-order) indices into LDS.
    if (t < cnt) {
#if HAS_ASYNC_LDS
      __attribute__((address_space(1))) int* g =
          (__attribute__((address_space(1))) int*)(idxbuf + (size_t)r * MAXP + t);
      __attribute__((address_space(3))) int* l =
          (__attribute__((address_space(3))) int*)(&shIdx[t]);
      __builtin_amdgcn_global_load_async_to_lds_b32(g, l, 0, 0);
      asm volatile("s_wait_asynccnt 0" ::: "memory");
#else
      shIdx[t] = idxbuf[(size_t)r * MAXP + t];
#endif
    }
    __syncthreads();
    if (t < cnt) {
      int v = shIdx[t];
      // deterministic position: rank of v among this voxel's point indices
      int pos = 0;
      for (int u = 0; u < cnt; u++) pos += (shIdx[u] < v) ? 1 : 0;
      float4 p = ((const float4*)pts)[v];
      ((float4*)outVox)[(size_t)r * MAXP + pos] = p;
    }
  } else {
    // Rare overflow (>64 pts in one voxel): deterministic block-wide rescan
    // keeping the 64 smallest original indices.
    int mylin = hKey[s];
    if (t == 0) shFound = 0;
    __syncthreads();
    for (int base = 0; base < n; base += MAXP) {
      int p = base + t;
      int f = (p < n && linA[p] == mylin) ? 1 : 0;
      shFlags[t] = f;
      __syncthreads();
      int pre = 0;
      for (int u = 0; u < t; u++) pre += shFlags[u];
      if (f) {
        int pos = shFound + pre;
        if (pos < MAXP) {
          float4 q = ((const float4*)pts)[p];
          ((float4*)outVox)[(size_t)r * MAXP + pos] = q;
        }
      }
      __syncthreads();
      if (t == MAXP - 1) shFound += pre + f;
      __syncthreads();
      if (shFound >= MAXP) break;
    }
  }
}

// ---------------- host-side launcher ----------------
extern "C" void kernel_launch(void* const* d_in, const int* in_sizes, int n_in,
                              void* d_out, int out_size, void* d_ws,
                              size_t ws_size, hipStream_t stream) {
  (void)n_in;
  (void)ws_size;
  const float* pts = (const float*)d_in[0];
  int n = in_sizes[0] / 4;
  int nAl = (n + 3) & ~3;  // keep int4 alignment for arrays after per-point data
  float* out = (float*)d_out;

  // workspace layout (ints)
  int* ws = (int*)d_ws;
  int* hKey = ws;
  int* hMin = hKey + HSIZE;
  int* hCnt = hMin + HSIZE;
  int* hRank = hCnt + HSIZE;
  int* linA = hRank + HSIZE;
  int* slotA = linA + nAl;
  int* arrival = slotA + nAl;
  int* rankSlot = arrival + MAXV;
  int* idxbuf = rankSlot + MAXV;
  int* misc = idxbuf + (size_t)MAXV * MAXP;  // [0]=voxel_num, [1..]=block sums
  int* bsums = misc + 1;

  // output layout (floats): voxels | coors | npv | voxel_num
  float* outVox = out;
  float* outCoors = out + (size_t)MAXV * MAXP * 4;
  float* outNpv = outCoors + (size_t)MAXV * 3;
  float* outVn = outNpv + MAXV;

  int nblocks = (n + SCAN_CHUNK - 1) / SCAN_CHUNK;

  vox_zero_f<<<4096, 256, 0, stream>>>(out, out_size);
  vox_init_ws<<<((HSIZE / 4) + 255) / 256, 256, 0, stream>>>(
      (int4*)hKey, (int4*)hMin, (int4*)hCnt, (int4*)hRank, (int4*)arrival);
  vox_assign<<<(n + 255) / 256, 256, 0, stream>>>(pts, n, hKey, hMin, hCnt,
                                                  linA, slotA);
  vox_blocksum<<<nblocks, SCAN_T, 0, stream>>>(slotA, hMin, n, bsums);
  vox_scan<<<1, 1, 0, stream>>>(nblocks, bsums, misc, outVn);
  vox_rank<<<nblocks, SCAN_T, 0, stream>>>(slotA, linA, hMin, hCnt, n, bsums,
                                           hRank, rankSlot, outCoors, outNpv);
  vox_collect<<<(n + 255) / 256, 256, 0, stream>>>(slotA, hRank, n, arrival,
                                                   idxbuf);
  vox_emit<<<MAXV, MAXP, 0, stream>>>(pts, n, idxbuf, rankSlot, hCnt, hKey,
                                      linA, misc, outVox);
}